// GCN_38560216383778
// MI455X (gfx1250) — compile-verified
//
#include <hip/hip_runtime.h>
#include <hip/hip_bf16.h>

typedef __attribute__((ext_vector_type(2))) float v2f;
typedef __attribute__((ext_vector_type(8))) float v8f;

#define N_NODES 100000

// ---------------------------------------------------------------------------
// Dense GEMM via V_WMMA_F32_16X16X4_F32. One wave (32 lanes) per 16x16 tile.
// M = 100000 (6250 exact tiles), N in {96,64,32,16}, K in {64,96,64,32}.
// A fragment: lane l holds X[Mbase + l%16][k0 + (l/16)*2 + {0,1}]  (float2)
// B fragment: lane l holds W[k0 + (l/16)*2 + {0,1}][Nbase + l%16]
// D store   : VGPR v -> row Mbase + (l/16)*8 + v, col Nbase + l%16
// ---------------------------------------------------------------------------
__global__ __launch_bounds__(32) void gcn_gemm_wmma(
    const float* __restrict__ X, const float* __restrict__ W,
    float* __restrict__ H, int K, int N)
{
    const int lane  = threadIdx.x;      // 0..31
    const int mrow  = lane & 15;        // 0..15
    const int khalf = lane >> 4;        // 0 or 1
    const int Mbase = blockIdx.x * 16;
    const int Nbase = blockIdx.y * 16;

    v8f acc = {};
    const float* __restrict__ xrow = X + (size_t)(Mbase + mrow) * K;
    const int col = Nbase + mrow;

    for (int k0 = 0; k0 < K; k0 += 4) {
        const int ka = k0 + khalf * 2;
        v2f a;
        a.x = xrow[ka];
        a.y = xrow[ka + 1];
        v2f b;
        b.x = W[(size_t)ka * N + col];
        b.y = W[(size_t)(ka + 1) * N + col];
        acc = __builtin_amdgcn_wmma_f32_16x16x4_f32(
            /*neg_a=*/false, a, /*neg_b=*/false, b,
            /*c_mod=*/(short)0, acc, /*reuse_a=*/false, /*reuse_b=*/false);
    }

    const int rbase = Mbase + khalf * 8;
#pragma unroll
    for (int v = 0; v < 8; ++v) {
        H[(size_t)(rbase + v) * N + col] = acc[v];
    }
}

// ---------------------------------------------------------------------------
// Degree pipeline: zero -> float-atomic count on dst -> dis = rsqrt(deg+1)
// ---------------------------------------------------------------------------
__global__ void gcn_zero(float* __restrict__ p, int n) {
    int t = blockIdx.x * blockDim.x + threadIdx.x;
    if (t < n) p[t] = 0.0f;
}

__global__ void gcn_deg_count(const int* __restrict__ dst, float* __restrict__ deg, int E) {
    int t = blockIdx.x * blockDim.x + threadIdx.x;
    if (t < E) atomicAdd(&deg[dst[t]], 1.0f);
}

__global__ void gcn_deg_finish(float* __restrict__ deg, int n) {
    int t = blockIdx.x * blockDim.x + threadIdx.x;
    if (t < n) deg[t] = rsqrtf(deg[t] + 1.0f);   // in place: deg -> deg_inv_sqrt
}

// ---------------------------------------------------------------------------
// agg[i,f] = h[i,f] * dis[i]^2 + b[f]   (full overwrite => accumulator init)
// ---------------------------------------------------------------------------
__global__ void gcn_selfloop_bias(const float* __restrict__ H,
                                  const float* __restrict__ dis,
                                  const float* __restrict__ bias,
                                  float* __restrict__ AGG, int F)
{
    int t = blockIdx.x * blockDim.x + threadIdx.x;
    long long total = (long long)N_NODES * F;
    if (t >= total) return;
    int i = t / F;
    int f = t - i * F;
    float d = dis[i];
    AGG[t] = H[t] * d * d + bias[f];
}

// ---------------------------------------------------------------------------
// Edge scatter: per (edge, float4 feature group) thread.
// Consecutive threads share an edge -> contiguous gather of h[src] row;
// atomics land in L2 (h/agg fully L2-resident at <=38MB).
// ---------------------------------------------------------------------------
__global__ void gcn_edge_scatter(const float* __restrict__ H,
                                 const int* __restrict__ src,
                                 const int* __restrict__ dst,
                                 const float* __restrict__ dis,
                                 float* __restrict__ AGG,
                                 int E, int F)
{
    int t = blockIdx.x * blockDim.x + threadIdx.x;
    int fg = F >> 2;                       // float4 groups per row
    if (t >= E * fg) return;
    int e = t / fg;
    int g = t - e * fg;
    int s = src[e];
    int d = dst[e];
    float norm = dis[s] * dis[d];
    const float4 hv = *((const float4*)(H + (size_t)s * F) + g);
    float* ap = AGG + (size_t)d * F + (size_t)g * 4;
    atomicAdd(ap + 0, hv.x * norm);
    atomicAdd(ap + 1, hv.y * norm);
    atomicAdd(ap + 2, hv.z * norm);
    atomicAdd(ap + 3, hv.w * norm);
}

__global__ void gcn_relu(float* __restrict__ p, long long n) {
    long long t = (long long)blockIdx.x * blockDim.x + threadIdx.x;
    if (t < n) p[t] = fmaxf(p[t], 0.0f);
}

// ---------------------------------------------------------------------------
// Driver
// ---------------------------------------------------------------------------
static void run_layer(const float* X, const float* W, const float* B,
                      float* H, float* AGG, const float* dis,
                      const int* src, const int* dst, int E,
                      int Fin, int Fout, bool do_relu, hipStream_t stream)
{
    dim3 gg(N_NODES / 16, Fout / 16);
    gcn_gemm_wmma<<<gg, 32, 0, stream>>>(X, W, H, Fin, Fout);

    long long nf = (long long)N_NODES * Fout;
    int blk = 256;
    gcn_selfloop_bias<<<(int)((nf + blk - 1) / blk), blk, 0, stream>>>(H, dis, B, AGG, Fout);

    long long et = (long long)E * (Fout >> 2);
    gcn_edge_scatter<<<(int)((et + blk - 1) / blk), blk, 0, stream>>>(H, src, dst, dis, AGG, E, Fout);

    if (do_relu)
        gcn_relu<<<(int)((nf + blk - 1) / blk), blk, 0, stream>>>(AGG, nf);
}

extern "C" void kernel_launch(void* const* d_in, const int* in_sizes, int n_in,
                              void* d_out, int out_size, void* d_ws, size_t ws_size,
                              hipStream_t stream)
{
    const float* x   = (const float*)d_in[0];
    const int*   ei  = (const int*)d_in[1];
    const float* W1  = (const float*)d_in[2];
    const float* b1  = (const float*)d_in[3];
    const float* W2  = (const float*)d_in[4];
    const float* b2  = (const float*)d_in[5];
    const float* W3  = (const float*)d_in[6];
    const float* b3  = (const float*)d_in[7];
    const float* W4  = (const float*)d_in[8];
    const float* b4  = (const float*)d_in[9];
    float* out = (float*)d_out;

    const int E = in_sizes[1] / 2;
    const int* src = ei;
    const int* dst = ei + E;

    // Workspace layout (floats): dis | H | bufA | bufB
    const size_t MAXF = 96;
    float* dis  = (float*)d_ws;
    float* H    = dis + N_NODES;
    float* bufA = H + (size_t)N_NODES * MAXF;
    float* bufB = bufA + (size_t)N_NODES * MAXF;

    int blk = 256;
    // degrees (recomputed every call; zero first since ws is not re-poisoned)
    gcn_zero<<<(N_NODES + blk - 1) / blk, blk, 0, stream>>>(dis, N_NODES);
    gcn_deg_count<<<(E + blk - 1) / blk, blk, 0, stream>>>(dst, dis, E);
    gcn_deg_finish<<<(N_NODES + blk - 1) / blk, blk, 0, stream>>>(dis, N_NODES);

    // Layer 1: 64 -> 96, relu, agg in bufA
    run_layer(x,    W1, b1, H, bufA, dis, src, dst, E, 64, 96, true,  stream);
    // Layer 2: 96 -> 64, relu, agg in bufB
    run_layer(bufA, W2, b2, H, bufB, dis, src, dst, E, 96, 64, true,  stream);
    // Layer 3: 64 -> 32, relu, agg in bufA
    run_layer(bufB, W3, b3, H, bufA, dis, src, dst, E, 64, 32, true,  stream);
    // Layer 4: 32 -> 16, no relu, agg written straight to d_out
    run_layer(bufA, W4, b4, H, out,  dis, src, dst, E, 32, 16, false, stream);

    (void)n_in; (void)out_size; (void)ws_size;
}